// GCNPredictor_74165495267521
// MI455X (gfx1250) — compile-verified
//
#include <hip/hip_runtime.h>
#include <hip/hip_bf16.h>

typedef __attribute__((ext_vector_type(2))) float v2f;
typedef __attribute__((ext_vector_type(8))) float v8f;

#define NCLS   20
#define DIN    2048
#define SIMD_  512
#define NNODE  1024
#define NIMG   4
#define NTOT   (NIMG * NNODE)   // 4096

// ---------------------------------------------------------------------------
// fp32 WMMA GEMM with compile-time shapes:  C[M,N] = A[M,K] @ B[K,N] (+ bias[N])
// One wave32 per 16x(16*NT) output strip: A fragment reused across NT wmmas.
// Out-of-range columns are clamped (their D columns are never stored), so the
// hot loop has no divergence and all strides are immediates.
// ---------------------------------------------------------------------------
template <int K, int N, int NT>
__global__ void wmma_gemm_bias(const float* __restrict__ A,
                               const float* __restrict__ B,
                               const float* __restrict__ bias,
                               float* __restrict__ C) {
    const int lane = threadIdx.x;            // 0..31
    const int m    = lane & 15;
    const int kh   = lane >> 4;              // K sub-half (0/1)
    const int row  = blockIdx.x * 16 + m;    // A row for this lane
    const int col0 = blockIdx.y * (16 * NT) + m;

    const float* __restrict__ Arow = A + (size_t)row * K + 2 * kh;
    const float* __restrict__ Bc[NT];
#pragma unroll
    for (int t = 0; t < NT; ++t) {
        int c = col0 + t * 16;
        if (c >= N) c = N - 1;               // clamp: garbage confined to unstored cols
        Bc[t] = B + (size_t)(2 * kh) * N + c;
    }

    v8f acc[NT];
#pragma unroll
    for (int t = 0; t < NT; ++t) acc[t] = (v8f){};

#pragma unroll 8
    for (int k = 0; k < K; k += 4) {
        const v2f a = *(const v2f*)(Arow + k);   // b64 load
#pragma unroll
        for (int t = 0; t < NT; ++t) {
            v2f b;
            b.x = Bc[t][(size_t)k * N];          // immediate-offset b32 loads
            b.y = Bc[t][(size_t)k * N + N];
            acc[t] = __builtin_amdgcn_wmma_f32_16x16x4_f32(
                false, a, false, b, (short)0, acc[t], false, false);
        }
    }

#pragma unroll
    for (int t = 0; t < NT; ++t) {
        const int c = col0 + t * 16;
        if (c < N) {
            const float bv = bias ? bias[c] : 0.0f;
#pragma unroll
            for (int r = 0; r < 8; ++r) {
                const int orow = blockIdx.x * 16 + r + 8 * kh;   // C/D VGPR layout
                C[(size_t)orow * N + c] = acc[t][r] + bv;
            }
        }
    }
}

// ---------------------------------------------------------------------------
// Per-image Gram matrix G = SF_b @ SF_b^T with fused threshold -> sim mask.
// grid (64, 16, 4), block 32; 4 column tiles per wave (A fragment reused).
// Both operands are rows of SF -> contiguous b64 loads.
// ---------------------------------------------------------------------------
__global__ void gram_mask_kernel(const float* __restrict__ SF,
                                 unsigned char* __restrict__ mask) {
    const int b    = blockIdx.z;
    const float* __restrict__ S = SF + (size_t)b * NNODE * SIMD_;
    const int lane = threadIdx.x;
    const int m    = lane & 15;
    const int kh   = lane >> 4;
    const int rowA = blockIdx.x * 16 + m;
    const int col0 = blockIdx.y * 64 + m;

    const float* __restrict__ Ap = S + (size_t)rowA * SIMD_ + 2 * kh;
    const float* __restrict__ Bp[4];
#pragma unroll
    for (int t = 0; t < 4; ++t)
        Bp[t] = S + (size_t)(col0 + t * 16) * SIMD_ + 2 * kh;   // B[k][n] = SF[n][k]

    v8f acc[4];
#pragma unroll
    for (int t = 0; t < 4; ++t) acc[t] = (v8f){};

#pragma unroll 8
    for (int k = 0; k < SIMD_; k += 4) {
        const v2f a = *(const v2f*)(Ap + k);
#pragma unroll
        for (int t = 0; t < 4; ++t) {
            const v2f bb = *(const v2f*)(Bp[t] + k);
            acc[t] = __builtin_amdgcn_wmma_f32_16x16x4_f32(
                false, a, false, bb, (short)0, acc[t], false, false);
        }
    }

    unsigned char* __restrict__ mk = mask + ((size_t)b << 20);
#pragma unroll
    for (int t = 0; t < 4; ++t) {
        const int col = col0 + t * 16;
#pragma unroll
        for (int r = 0; r < 8; ++r) {
            const int row = blockIdx.x * 16 + r + 8 * kh;
            mk[(size_t)row * NNODE + col] =
                (unsigned char)((acc[t][r] >= 0.4f) || (row == col));
        }
    }
}

// ---------------------------------------------------------------------------
// IoU mask per image: mask[b][i][j] = (IoU >= 0.4) | (i==j). 4*1024*1024 threads.
// ---------------------------------------------------------------------------
__global__ void iou_mask_kernel(const float* __restrict__ boxes,
                                unsigned char* __restrict__ mask) {
    const size_t idx = (size_t)blockIdx.x * 256 + threadIdx.x;   // < 4M
    const int b = (int)(idx >> 20);
    const int i = (int)((idx >> 10) & 1023);
    const int j = (int)(idx & 1023);
    const float* bi = boxes + ((size_t)b * NNODE + i) * 4;
    const float* bj = boxes + ((size_t)b * NNODE + j) * 4;
    const float ax0 = bi[0], ay0 = bi[1], ax1 = bi[2], ay1 = bi[3];
    const float bx0 = bj[0], by0 = bj[1], bx1 = bj[2], by1 = bj[3];
    const float areaA = (ax1 - ax0) * (ay1 - ay0);
    const float areaB = (bx1 - bx0) * (by1 - by0);
    const float lx = fmaxf(ax0, bx0), ly = fmaxf(ay0, by0);
    const float rx = fminf(ax1, bx1), ry = fminf(ay1, by1);
    const float w = fmaxf(rx - lx, 0.f), h = fmaxf(ry - ly, 0.f);
    const float inter = w * h;
    const float iou = inter / (areaA + areaB - inter + 1e-9f);
    mask[idx] = (unsigned char)((iou >= 0.4f) || (i == j));
}

// ---------------------------------------------------------------------------
// L2-normalize rows of SF [4096, 512]. One block (256 thr) per row.
// ---------------------------------------------------------------------------
__global__ void l2norm_rows_kernel(float* __restrict__ SF) {
    __shared__ float sd[256];
    const int t = threadIdx.x;
    float* __restrict__ p = SF + (size_t)blockIdx.x * SIMD_;
    const float a = p[t], b = p[t + 256];
    sd[t] = a * a + b * b;
    __syncthreads();
    for (int off = 128; off > 0; off >>= 1) {
        if (t < off) sd[t] += sd[t + off];
        __syncthreads();
    }
    const float inv = 1.0f / (sqrtf(sd[0]) + 1e-9f);
    p[t] = a * inv;
    p[t + 256] = b * inv;
}

// ---------------------------------------------------------------------------
// Attention coefficient vectors: es[s,h] = sum_f h[s,h,f]*a_src[h,f], same for ed.
// ---------------------------------------------------------------------------
__global__ void gat1_vec_kernel(const float* __restrict__ Hf,
                                const float* __restrict__ asrc,
                                const float* __restrict__ adst,
                                float* __restrict__ ES, float* __restrict__ ED) {
    const int idx = blockIdx.x * 256 + threadIdx.x;
    if (idx >= NTOT * 8) return;
    const int s = idx >> 3, h = idx & 7;
    const float* hp = Hf + (size_t)s * 64 + h * 8;
    float se = 0.f, de = 0.f;
#pragma unroll
    for (int f = 0; f < 8; ++f) {
        const float v = hp[f];
        se += v * asrc[h * 8 + f];
        de += v * adst[h * 8 + f];
    }
    ES[idx] = se;
    ED[idx] = de;
}

__global__ void gat2_vec_kernel(const float* __restrict__ Hf,
                                const float* __restrict__ asrc,
                                const float* __restrict__ adst,
                                float* __restrict__ ES, float* __restrict__ ED) {
    const int s = blockIdx.x * 256 + threadIdx.x;
    if (s >= NTOT) return;
    float se = 0.f, de = 0.f;
#pragma unroll
    for (int c = 0; c < NCLS; ++c) {
        const float v = Hf[(size_t)s * NCLS + c];
        se += v * asrc[c];
        de += v * adst[c];
    }
    ES[s] = se;
    ED[s] = de;
}

// ---------------------------------------------------------------------------
// Masked-softmax GAT aggregation. One block (256 thr) per destination node.
// out[d, h*F+f] = (sum_s w * h[s,h,f]) / (sum_s w) + bias,  w = exp(leaky(es+ed)-max)
// Masks are symmetric -> read mask row d (coalesced over s).
// ---------------------------------------------------------------------------
template <int H, int F>
__global__ void gat_agg_kernel(const float* __restrict__ hfeat,
                               const float* __restrict__ esv,
                               const float* __restrict__ edv,
                               const unsigned char* __restrict__ mask,
                               const float* __restrict__ bias,
                               float* __restrict__ out, int applyElu) {
    __shared__ float sd[256];
    const int d  = blockIdx.x;          // 0..4095
    const int t  = threadIdx.x;         // 0..255
    const int b  = d >> 10, dl = d & 1023;
    const unsigned char* __restrict__ mrow = mask + ((size_t)b << 20) + ((size_t)dl << 10);
    const int sbase = b << 10;

    for (int h = 0; h < H; ++h) {
        const float edd = edv[d * H + h];
        // pass 1: max over valid sources
        float lmax = -1e30f;
        for (int s = t; s < NNODE; s += 256) {
            if (mrow[s]) {
                float v = esv[(sbase + s) * H + h] + edd;
                v = (v > 0.f) ? v : 0.2f * v;          // leaky_relu 0.2
                lmax = fmaxf(lmax, v);
            }
        }
        sd[t] = lmax; __syncthreads();
        for (int off = 128; off > 0; off >>= 1) {
            if (t < off) sd[t] = fmaxf(sd[t], sd[t + off]);
            __syncthreads();
        }
        const float gmax = sd[0]; __syncthreads();

        // pass 2: weighted sums
        float sw = 0.f;
        float acc[F];
#pragma unroll
        for (int f = 0; f < F; ++f) acc[f] = 0.f;
        for (int s = t; s < NNODE; s += 256) {
            if (mrow[s]) {
                float v = esv[(sbase + s) * H + h] + edd;
                v = (v > 0.f) ? v : 0.2f * v;
                const float w = __expf(v - gmax);
                sw += w;
                const float* hp = hfeat + (size_t)(sbase + s) * (H * F) + h * F;
#pragma unroll
                for (int f = 0; f < F; ++f) acc[f] += w * hp[f];
            }
        }
        sd[t] = sw; __syncthreads();
        for (int off = 128; off > 0; off >>= 1) {
            if (t < off) sd[t] += sd[t + off];
            __syncthreads();
        }
        const float inv = 1.0f / sd[0]; __syncthreads();

        for (int f = 0; f < F; ++f) {
            sd[t] = acc[f]; __syncthreads();
            for (int off = 128; off > 0; off >>= 1) {
                if (t < off) sd[t] += sd[t + off];
                __syncthreads();
            }
            if (t == 0) {
                float o = sd[0] * inv + bias[h * F + f];
                if (applyElu) o = (o > 0.f) ? o : (__expf(o) - 1.0f);
                out[(size_t)d * (H * F) + h * F + f] = o;
            }
            __syncthreads();
        }
    }
}

// ---------------------------------------------------------------------------
// node_score = softmax(n2 @ node_W + node_b) per node. One thread per node.
// ---------------------------------------------------------------------------
__global__ void node_score_kernel(const float* __restrict__ N2,
                                  const float* __restrict__ W,
                                  const float* __restrict__ bb,
                                  float* __restrict__ node_out) {
    const int s = blockIdx.x * 256 + threadIdx.x;
    if (s >= NTOT) return;
    float tv[NCLS];
#pragma unroll
    for (int c = 0; c < NCLS; ++c) tv[c] = bb[c];
    for (int j = 0; j < NCLS; ++j) {
        const float v = N2[(size_t)s * NCLS + j];
#pragma unroll
        for (int c = 0; c < NCLS; ++c) tv[c] += v * W[j * NCLS + c];
    }
    float mx = tv[0];
#pragma unroll
    for (int c = 1; c < NCLS; ++c) mx = fmaxf(mx, tv[c]);
    float sum = 0.f;
#pragma unroll
    for (int c = 0; c < NCLS; ++c) { tv[c] = __expf(tv[c] - mx); sum += tv[c]; }
    const float inv = 1.0f / sum;
#pragma unroll
    for (int c = 0; c < NCLS; ++c) node_out[(size_t)s * NCLS + c] = tv[c] * inv;
}

// ---------------------------------------------------------------------------
// graph_score[b,c] = mean over nodes of node_score (column sums of a_matrix are
// exactly 1 by softmax normalization, so the attention GEMM collapses).
// Deterministic sequential reduction, no atomics.
// ---------------------------------------------------------------------------
__global__ void graph_score_kernel(const float* __restrict__ node,
                                   float* __restrict__ g) {
    const int t = threadIdx.x;
    if (t >= NIMG * NCLS) return;
    const int b = t / NCLS, c = t % NCLS;
    float sum = 0.f;
    for (int d = 0; d < NNODE; ++d)
        sum += node[((size_t)(b << 10) + d) * NCLS + c];
    g[t] = sum * (1.0f / (float)NNODE);
}

// ---------------------------------------------------------------------------
extern "C" void kernel_launch(void* const* d_in, const int* in_sizes, int n_in,
                              void* d_out, int out_size, void* d_ws, size_t ws_size,
                              hipStream_t stream) {
    (void)in_sizes; (void)n_in; (void)out_size; (void)ws_size;
    const float* x1    = (const float*)d_in[0];   // [4096, 2048]
    const float* boxes = (const float*)d_in[1];   // [4, 1024, 4]
    const float* simW  = (const float*)d_in[2];   // [2048, 512]
    const float* simB  = (const float*)d_in[3];   // [512]
    const float* g1W   = (const float*)d_in[4];   // [2048, 64]
    const float* g1as  = (const float*)d_in[5];   // [8, 8]
    const float* g1ad  = (const float*)d_in[6];   // [8, 8]
    const float* g1b   = (const float*)d_in[7];   // [64]
    const float* g2W   = (const float*)d_in[8];   // [64, 20]
    const float* g2as  = (const float*)d_in[9];   // [1, 20]
    const float* g2ad  = (const float*)d_in[10];  // [1, 20]
    const float* g2b   = (const float*)d_in[11];  // [20]
    const float* nW    = (const float*)d_in[12];  // [20, 20]
    const float* nb    = (const float*)d_in[13];  // [20]
    const float* refW  = (const float*)d_in[14];  // [3, 2048, 20]
    const float* refB  = (const float*)d_in[15];  // [3, 20]
    const float* bbW   = (const float*)d_in[16];  // [3, 2048, 80]
    const float* bbB   = (const float*)d_in[17];  // [3, 80]
    float* out = (float*)d_out;
    // d_out layout: graph[80] | node[81920] | ref[245760] | bbox[983040]
    float* out_node = out + 80;
    float* out_ref  = out + 82000;
    float* out_bbox = out + 327760;

    // workspace layout (floats then byte masks); ~19.9 MB total
    float* ws  = (float*)d_ws;
    float* SF  = ws;                 // 4096*512
    float* H1f = SF  + 2097152;      // 4096*64
    float* ES1 = H1f + 262144;       // 4096*8
    float* ED1 = ES1 + 32768;        // 4096*8
    float* N1  = ED1 + 32768;        // 4096*64
    float* H2f = N1  + 262144;       // 4096*20
    float* ES2 = H2f + 81920;        // 4096
    float* ED2 = ES2 + 4096;         // 4096
    float* N2  = ED2 + 4096;         // 4096*20
    unsigned char* iouM = (unsigned char*)(N2 + 81920);        // 4 MB
    unsigned char* simM = iouM + (size_t)NIMG * NNODE * NNODE; // 4 MB

    // 1. IoU adjacency masks (symmetric)
    iou_mask_kernel<<<(NIMG * NNODE * NNODE) / 256, 256, 0, stream>>>(boxes, iouM);

    // 2. Sim features: SF = x1 @ sim_W + sim_b, then row L2 normalize
    wmma_gemm_bias<DIN, SIMD_, 4><<<dim3(NTOT / 16, 8), 32, 0, stream>>>(x1, simW, simB, SF);
    l2norm_rows_kernel<<<NTOT, 256, 0, stream>>>(SF);

    // 3. Sim masks: threshold(SF_b @ SF_b^T) per image (symmetric)
    gram_mask_kernel<<<dim3(NNODE / 16, 16, NIMG), 32, 0, stream>>>(SF, simM);

    // 4. GAT1: h = x1 @ gat1_W; attention scalars; masked-softmax aggregation + bias + ELU
    wmma_gemm_bias<DIN, 64, 4><<<dim3(NTOT / 16, 1), 32, 0, stream>>>(x1, g1W, nullptr, H1f);
    gat1_vec_kernel<<<(NTOT * 8) / 256, 256, 0, stream>>>(H1f, g1as, g1ad, ES1, ED1);
    gat_agg_kernel<8, 8><<<NTOT, 256, 0, stream>>>(H1f, ES1, ED1, iouM, g1b, N1, 1);

    // 5. GAT2: h2 = n1 @ gat2_W; aggregation over sim mask + bias (no activation)
    wmma_gemm_bias<64, NCLS, 2><<<dim3(NTOT / 16, 1), 32, 0, stream>>>(N1, g2W, nullptr, H2f);
    gat2_vec_kernel<<<NTOT / 256, 256, 0, stream>>>(H2f, g2as, g2ad, ES2, ED2);
    gat_agg_kernel<1, NCLS><<<NTOT, 256, 0, stream>>>(H2f, ES2, ED2, simM, g2b, N2, 0);

    // 6. node_score = softmax(n2 @ node_W + node_b); graph_score = mean(node_score)
    node_score_kernel<<<NTOT / 256, 256, 0, stream>>>(N2, nW, nb, out_node);
    graph_score_kernel<<<1, 128, 0, stream>>>(out_node, out);

    // 7. Refinement heads: ref_logits[k] = x1 @ ref_W[k] + ref_b[k]   (N=20)
    for (int k = 0; k < 3; ++k) {
        wmma_gemm_bias<DIN, NCLS, 2><<<dim3(NTOT / 16, 1), 32, 0, stream>>>(
            x1, refW + (size_t)k * DIN * NCLS, refB + k * NCLS,
            out_ref + (size_t)k * NTOT * NCLS);
    }
    // 8. BBox heads: bbox_preds[k] = x1 @ bbox_W[k] + bbox_b[k]       (N=80)
    for (int k = 0; k < 3; ++k) {
        wmma_gemm_bias<DIN, 80, 5><<<dim3(NTOT / 16, 1), 32, 0, stream>>>(
            x1, bbW + (size_t)k * DIN * 80, bbB + k * 80,
            out_bbox + (size_t)k * NTOT * 80);
    }
}